// GraphTransformerLayer_24558622998902
// MI455X (gfx1250) — compile-verified
//
#include <hip/hip_runtime.h>
#include <math.h>

// Problem constants (match reference)
#define NNODE 20000
#define FDIM  512
#define HEADS 8
#define HDIM  64
#define NEDGE 320000

typedef __attribute__((ext_vector_type(2))) float v2f;
typedef __attribute__((ext_vector_type(4))) float v4f;
typedef __attribute__((ext_vector_type(8))) float v8f;

// ---------------------------------------------------------------------------
// utility: float atomic max via sign-split int/uint atomics
// ---------------------------------------------------------------------------
__device__ __forceinline__ void atomicMaxFloat(float* addr, float value) {
    if (value >= 0.0f)
        atomicMax((int*)addr, __float_as_int(value));
    else
        atomicMin((unsigned int*)addr, __float_as_uint(value));
}

__global__ void fill_kernel(float* __restrict__ p, float val, size_t n) {
    size_t i = (size_t)blockIdx.x * blockDim.x + threadIdx.x;
    if (i < n) p[i] = val;
}

// ---------------------------------------------------------------------------
// FP32 WMMA GEMM: C[M,Nc] = A[M,K] @ B[K,Nc]  (+ epilogue)
// epilogue: 0 = none, 1 = relu, 2 = C = Res + A@B
//
// Block tile 64x128, 256 threads = 8 waves arranged 2(M) x 4(N); each wave
// computes a 32x32 sub-tile as a 2x2 grid of 16x16 WMMA fragments
// (32 v_wmma per K-panel between barriers).
//
// LDS layouts store K in packed pairs [k/2][row][2] so each 16x4 fp32 WMMA
// fragment operand (per-lane K pair: lanes 0-15 -> K+0/K+1, lanes 16-31 ->
// K+2/K+3, ISA 7.12.2) is ONE contiguous 8B LDS load into an even VGPR pair.
// ---------------------------------------------------------------------------
#define MT  64
#define NTB 128
#define KT  32

__global__ __launch_bounds__(256) void gemm_wmma_f32(
    const float* __restrict__ A, const float* __restrict__ B,
    const float* __restrict__ Res, float* __restrict__ C,
    int M, int K, int Nc, int epilogue)
{
    __shared__ float As[KT / 2][MT + 2][2];    // (k,m) at As[k>>1][m][k&1]
    __shared__ float Bs[KT / 2][NTB + 2][2];   // (k,n) at Bs[k>>1][n][k&1]

    const int tid  = threadIdx.x;
    const int lane = tid & 31;
    const int wave = tid >> 5;
    const int wm   = wave >> 2;   // 0..1 : 32-row slice
    const int wn   = wave & 3;    // 0..3 : 32-col slice
    const int row0 = blockIdx.x * MT;
    const int col0 = blockIdx.y * NTB;

    v8f c00 = {}; v8f c01 = {}; v8f c10 = {}; v8f c11 = {};

    const int l15  = lane & 15;
    const int half = lane >> 4;              // 0 or 1 (K sub-pair per half-wave)
    const int ma0  = (wm << 5) + l15;        // A-frag rows (frag 0)
    const int ma1  = ma0 + 16;               // A-frag rows (frag 1)
    const int nb0  = (wn << 5) + l15;        // B-frag cols (frag 0)
    const int nb1  = nb0 + 16;               // B-frag cols (frag 1)

    for (int k0 = 0; k0 < K; k0 += KT) {
        // stage A: 64 rows x 32 k (512 float4), transposed into packed pairs
        #pragma unroll
        for (int it = 0; it < 2; ++it) {
            int idx = tid + it * 256;          // 0..511
            int m   = idx >> 3;                // 8 float4 per row
            int kk  = (idx & 7) << 2;
            int grow = row0 + m;
            v4f a;
            if (grow < M) a = *(const v4f*)(A + (size_t)grow * K + k0 + kk);
            else          a = (v4f){0.f, 0.f, 0.f, 0.f};
            int p = kk >> 1;
            *(v2f*)(&As[p + 0][m][0]) = (v2f){a.x, a.y};
            *(v2f*)(&As[p + 1][m][0]) = (v2f){a.z, a.w};
        }
        // stage B: 32 k x 128 n (1024 float4) into packed pairs
        #pragma unroll
        for (int it = 0; it < 4; ++it) {
            int idx = tid + it * 256;          // 0..1023
            int kk  = idx >> 5;                // 32 float4 per k-row
            int n   = (idx & 31) << 2;
            v4f b = *(const v4f*)(B + (size_t)(k0 + kk) * Nc + col0 + n);
            int p = kk >> 1, sub = kk & 1;
            Bs[p][n + 0][sub] = b.x; Bs[p][n + 1][sub] = b.y;
            Bs[p][n + 2][sub] = b.z; Bs[p][n + 3][sub] = b.w;
        }
        __syncthreads();

        #pragma unroll
        for (int kk = 0; kk < KT; kk += 4) {
            int p = (kk >> 1) + half;          // lane's K pair index
            v2f a0 = *(const v2f*)(&As[p][ma0][0]);
            v2f a1 = *(const v2f*)(&As[p][ma1][0]);
            v2f b0 = *(const v2f*)(&Bs[p][nb0][0]);
            v2f b1 = *(const v2f*)(&Bs[p][nb1][0]);
            c00 = __builtin_amdgcn_wmma_f32_16x16x4_f32(false, a0, false, b0, (short)0, c00, false, false);
            c01 = __builtin_amdgcn_wmma_f32_16x16x4_f32(false, a0, false, b1, (short)0, c01, false, false);
            c10 = __builtin_amdgcn_wmma_f32_16x16x4_f32(false, a1, false, b0, (short)0, c10, false, false);
            c11 = __builtin_amdgcn_wmma_f32_16x16x4_f32(false, a1, false, b1, (short)0, c11, false, false);
        }
        __syncthreads();
    }

    // store: VGPR j -> row j (lanes 0-15) / row j+8 (lanes 16-31), col = lane&15
    const int rb0 = row0 + (wm << 5) + (half << 3);   // rows for c0x
    const int rb1 = rb0 + 16;                         // rows for c1x
    const int cb0 = col0 + (wn << 5) + l15;
    #pragma unroll
    for (int j = 0; j < 8; ++j) {
        int r0 = rb0 + j, r1 = rb1 + j;
        if (r0 < M) {
            size_t o0 = (size_t)r0 * Nc + cb0;
            float x0 = c00[j], x1 = c01[j];
            if (epilogue == 1)      { x0 = fmaxf(x0, 0.f); x1 = fmaxf(x1, 0.f); }
            else if (epilogue == 2) { x0 += Res[o0];       x1 += Res[o0 + 16];  }
            C[o0] = x0; C[o0 + 16] = x1;
        }
        if (r1 < M) {
            size_t o1 = (size_t)r1 * Nc + cb0;
            float x0 = c10[j], x1 = c11[j];
            if (epilogue == 1)      { x0 = fmaxf(x0, 0.f); x1 = fmaxf(x1, 0.f); }
            else if (epilogue == 2) { x0 += Res[o1];       x1 += Res[o1 + 16];  }
            C[o1] = x0; C[o1 + 16] = x1;
        }
    }
}

// ---------------------------------------------------------------------------
// Edge logits: one wave per edge. 4 lanes per head x 8 heads.
// logit = clip(q[src,h,:].k[dst,h,:], -5, 5) * 8  (= /SCALING, SCALING=1/8)
// also segment-max into m[dst,h]
// ---------------------------------------------------------------------------
__global__ __launch_bounds__(256) void edge_logits_kernel(
    const float* __restrict__ q, const float* __restrict__ kmat,
    const int* __restrict__ esrc, const int* __restrict__ edst,
    float* __restrict__ logits, float* __restrict__ m, int E)
{
    int e = blockIdx.x * (blockDim.x >> 5) + (threadIdx.x >> 5);
    int lane = threadIdx.x & 31;
    if (e >= E) return;
    int s = esrc[e], d = edst[e];
    int h  = lane >> 2;
    int li = lane & 3;
    const float* qp = q    + (size_t)s * FDIM + h * HDIM + li * 16;
    const float* kp = kmat + (size_t)d * FDIM + h * HDIM + li * 16;
    float acc = 0.f;
    #pragma unroll
    for (int i = 0; i < 16; i += 4) {
        v4f qa = *(const v4f*)(qp + i);
        v4f ka = *(const v4f*)(kp + i);
        acc += qa.x * ka.x + qa.y * ka.y + qa.z * ka.z + qa.w * ka.w;
    }
    acc += __shfl_xor(acc, 1, 32);
    acc += __shfl_xor(acc, 2, 32);
    if (li == 0) {
        float lg = fminf(fmaxf(acc, -5.0f), 5.0f) * 8.0f;
        logits[(size_t)e * HEADS + h] = lg;
        atomicMaxFloat(m + (size_t)d * HEADS + h, lg);
    }
}

// ex = exp(logit - m[dst]) written in place over logits; denom[dst] += ex
__global__ void edge_exp_kernel(const int* __restrict__ edst,
                                float* __restrict__ logits,
                                const float* __restrict__ m,
                                float* __restrict__ denom, int E)
{
    size_t i = (size_t)blockIdx.x * blockDim.x + threadIdx.x;
    if (i >= (size_t)E * HEADS) return;
    int e = (int)(i >> 3);
    int h = (int)(i & 7);
    int d = edst[e];
    float ex = expf(logits[i] - m[(size_t)d * HEADS + h]);
    logits[i] = ex;
    atomicAdd(denom + (size_t)d * HEADS + h, ex);
}

// agg[dst] += v[src] * sa  (one wave per edge, coalesced 32-float rows)
__global__ __launch_bounds__(256) void edge_agg_kernel(
    const float* __restrict__ vmat, const float* __restrict__ ex,
    const int* __restrict__ esrc, const int* __restrict__ edst,
    const float* __restrict__ denom, float* __restrict__ agg, int E)
{
    int e = blockIdx.x * (blockDim.x >> 5) + (threadIdx.x >> 5);
    int lane = threadIdx.x & 31;
    if (e >= E) return;
    int s = esrc[e], d = edst[e];
    float sa = 0.f;
    if (lane < HEADS)
        sa = ex[(size_t)e * HEADS + lane] / denom[(size_t)d * HEADS + lane];
    const float* vp = vmat + (size_t)s * FDIM;
    float* ap = agg + (size_t)d * FDIM;
    #pragma unroll
    for (int i = 0; i < 16; ++i) {
        int c = i * 32 + lane;
        float w = __shfl(sa, c >> 6, 32);
        atomicAdd(ap + c, vp[c] * w);
    }
}

// ---------------------------------------------------------------------------
// fused gate + blend + LayerNorm1 : one block (256 thr) per node
// g = sigmoid(sum rst*Wg0 + skip*Wg1 + (rst-skip)*Wg2)
// h = LN(rst*g + skip*(1-g)) * g1 + b1
// ---------------------------------------------------------------------------
__global__ __launch_bounds__(256) void gate_ln1_kernel(
    const float* __restrict__ rst, const float* __restrict__ skip,
    const float* __restrict__ Wg, const float* __restrict__ g1,
    const float* __restrict__ b1, float* __restrict__ hout)
{
    int n = blockIdx.x, t = threadIdx.x;
    __shared__ float red[256];
    __shared__ float ybuf[FDIM];
    const float* rp = rst  + (size_t)n * FDIM;
    const float* sp = skip + (size_t)n * FDIM;

    float acc = 0.f;
    for (int f = t; f < FDIM; f += 256) {
        float r = rp[f], s = sp[f];
        acc += r * Wg[f] + s * Wg[FDIM + f] + (r - s) * Wg[2 * FDIM + f];
    }
    red[t] = acc; __syncthreads();
    for (int st = 128; st > 0; st >>= 1) { if (t < st) red[t] += red[t + st]; __syncthreads(); }
    float g = 1.0f / (1.0f + expf(-red[0]));
    __syncthreads();

    float lsum = 0.f;
    for (int f = t; f < FDIM; f += 256) {
        float y = rp[f] * g + sp[f] * (1.0f - g);
        ybuf[f] = y; lsum += y;
    }
    red[t] = lsum; __syncthreads();
    for (int st = 128; st > 0; st >>= 1) { if (t < st) red[t] += red[t + st]; __syncthreads(); }
    float mu = red[0] / FDIM;
    __syncthreads();

    float lvar = 0.f;
    for (int f = t; f < FDIM; f += 256) { float d = ybuf[f] - mu; lvar += d * d; }
    red[t] = lvar; __syncthreads();
    for (int st = 128; st > 0; st >>= 1) { if (t < st) red[t] += red[t + st]; __syncthreads(); }
    float inv = rsqrtf(red[0] / FDIM + 1e-5f);

    for (int f = t; f < FDIM; f += 256)
        hout[(size_t)n * FDIM + f] = (ybuf[f] - mu) * inv * g1[f] + b1[f];
}

// plain LayerNorm (for ln2): one block per node
__global__ __launch_bounds__(256) void ln_kernel(
    const float* __restrict__ x, const float* __restrict__ g,
    const float* __restrict__ b, float* __restrict__ out)
{
    int n = blockIdx.x, t = threadIdx.x;
    __shared__ float red[256];
    const float* xp = x + (size_t)n * FDIM;

    float lsum = 0.f;
    for (int f = t; f < FDIM; f += 256) lsum += xp[f];
    red[t] = lsum; __syncthreads();
    for (int st = 128; st > 0; st >>= 1) { if (t < st) red[t] += red[t + st]; __syncthreads(); }
    float mu = red[0] / FDIM;
    __syncthreads();

    float lvar = 0.f;
    for (int f = t; f < FDIM; f += 256) { float d = xp[f] - mu; lvar += d * d; }
    red[t] = lvar; __syncthreads();
    for (int st = 128; st > 0; st >>= 1) { if (t < st) red[t] += red[t + st]; __syncthreads(); }
    float inv = rsqrtf(red[0] / FDIM + 1e-5f);

    for (int f = t; f < FDIM; f += 256)
        out[(size_t)n * FDIM + f] = (xp[f] - mu) * inv * g[f] + b[f];
}

// ---------------------------------------------------------------------------
extern "C" void kernel_launch(void* const* d_in, const int* in_sizes, int n_in,
                              void* d_out, int out_size, void* d_ws, size_t ws_size,
                              hipStream_t stream) {
    const float* feat  = (const float*)d_in[0];
    const int*   esrc  = (const int*)  d_in[1];
    const int*   edst  = (const int*)  d_in[2];
    const float* Wq    = (const float*)d_in[3];
    const float* Wk    = (const float*)d_in[4];
    const float* Wv    = (const float*)d_in[5];
    const float* Wnode = (const float*)d_in[6];
    const float* Wskip = (const float*)d_in[7];
    const float* Wgate = (const float*)d_in[8];
    const float* ln1g  = (const float*)d_in[9];
    const float* ln1b  = (const float*)d_in[10];
    const float* ln2g  = (const float*)d_in[11];
    const float* ln2b  = (const float*)d_in[12];
    const float* W1    = (const float*)d_in[13];
    const float* W2    = (const float*)d_in[14];

    const int Nn = in_sizes[0] / FDIM;   // 20000
    const int Ee = in_sizes[1];          // 320000
    const size_t NF = (size_t)Nn * FDIM;
    const size_t EH = (size_t)Ee * HEADS;
    const size_t NH = (size_t)Nn * HEADS;

    float* ws    = (float*)d_ws;
    float* qbuf  = ws;
    float* kbuf  = ws + NF;
    float* vbuf  = ws + 2 * NF;
    float* sbuf  = ws + 3 * NF;      // skip
    float* abuf  = ws + 4 * NF;      // agg
    float* exbuf = ws + 5 * NF;      // logits -> ex (in place)
    float* mbuf  = exbuf + EH;
    float* dbuf  = mbuf + NH;        // denom
    // slot reuse after lifetimes end:
    float* rst  = qbuf;              // q dead after edge_logits
    float* hbuf = kbuf;              // k dead after edge_logits
    float* ln2h = abuf;              // agg dead after rst GEMM
    float* ffn1 = vbuf;              // v dead after edge_agg
    float* out  = (float*)d_out;

    // init accumulators every call (graph-replay safe)
    fill_kernel<<<(unsigned)((NH + 255) / 256), 256, 0, stream>>>(mbuf, -INFINITY, NH);
    fill_kernel<<<(unsigned)((NH + 255) / 256), 256, 0, stream>>>(dbuf, 0.0f, NH);
    fill_kernel<<<(unsigned)((NF + 255) / 256), 256, 0, stream>>>(abuf, 0.0f, NF);

    dim3 ggrid((Nn + MT - 1) / MT, FDIM / NTB);
    gemm_wmma_f32<<<ggrid, 256, 0, stream>>>(feat, Wq,    nullptr, qbuf, Nn, FDIM, FDIM, 0);
    gemm_wmma_f32<<<ggrid, 256, 0, stream>>>(feat, Wk,    nullptr, kbuf, Nn, FDIM, FDIM, 0);
    gemm_wmma_f32<<<ggrid, 256, 0, stream>>>(feat, Wv,    nullptr, vbuf, Nn, FDIM, FDIM, 0);
    gemm_wmma_f32<<<ggrid, 256, 0, stream>>>(feat, Wskip, nullptr, sbuf, Nn, FDIM, FDIM, 0);

    int eblocks = (Ee + 7) / 8;   // 8 waves per 256-thread block, 1 wave/edge
    edge_logits_kernel<<<eblocks, 256, 0, stream>>>(qbuf, kbuf, esrc, edst, exbuf, mbuf, Ee);
    edge_exp_kernel<<<(unsigned)((EH + 255) / 256), 256, 0, stream>>>(edst, exbuf, mbuf, dbuf, Ee);
    edge_agg_kernel<<<eblocks, 256, 0, stream>>>(vbuf, exbuf, esrc, edst, dbuf, abuf, Ee);

    gemm_wmma_f32<<<ggrid, 256, 0, stream>>>(abuf, Wnode, nullptr, rst, Nn, FDIM, FDIM, 0);

    gate_ln1_kernel<<<Nn, 256, 0, stream>>>(rst, sbuf, Wgate, ln1g, ln1b, hbuf);
    ln_kernel<<<Nn, 256, 0, stream>>>(hbuf, ln2g, ln2b, ln2h);

    gemm_wmma_f32<<<ggrid, 256, 0, stream>>>(ln2h, W1, nullptr, ffn1, Nn, FDIM, FDIM, 1); // relu
    gemm_wmma_f32<<<ggrid, 256, 0, stream>>>(ffn1, W2, hbuf,    out,  Nn, FDIM, FDIM, 2); // +h
}